// DecMoE_19851338842797
// MI455X (gfx1250) — compile-verified
//
#include <hip/hip_runtime.h>
#include <math.h>

typedef float v2f __attribute__((ext_vector_type(2)));
typedef float v8f __attribute__((ext_vector_type(8)));
typedef int   vsi4 __attribute__((vector_size(16)));   // matches builtin param type

#define B_ROWS 262144
#define E_NUM 4
#define DS_DIM 128
#define C_DIM 128
#define H_DIM 32
#define OUT_Q 81
#define NPAD 96

#define ROWS_PER_BLOCK 64
#define WAVES_PER_BLOCK 4

// d_ws float layout
#define WS_G0 0
#define WS_G1 1
#define WS_E0 3              // int view
#define WS_E1 4              // int view
#define WS_W2P 8             // 2 * 32 * 96 floats
#define WS_B2P (8 + 2 * H_DIM * NPAD)  // 2 * 96 floats

__device__ __forceinline__ float cv_squared4(const float* x) {
    float mean = 0.25f * (x[0] + x[1] + x[2] + x[3]);
    float var = 0.f;
    for (int i = 0; i < 4; ++i) { float d = x[i] - mean; var += d * d; }
    var *= (1.0f / 3.0f);  // ddof=1
    return var / (mean * mean + 1e-10f);
}

// One small block: gating (identical for all rows since x_ds rows are tiled),
// loss scalar, and precompute of W2P = w2[e] @ P[e] (32x96 padded) and b2P.
__global__ void gate_kernel(const float* __restrict__ x_ds,
                            const float* __restrict__ w_gate,
                            const float* __restrict__ w2,
                            const float* __restrict__ b2,
                            float* __restrict__ wsf,
                            float* __restrict__ loss_out) {
    __shared__ int sh_e[2];
    int tid = threadIdx.x;
    if (tid == 0) {
        float logits[E_NUM];
        for (int e = 0; e < E_NUM; ++e) {
            float s = 0.f;
            for (int d = 0; d < DS_DIM; ++d) s += x_ds[d] * w_gate[d * E_NUM + e];
            logits[e] = s;
        }
        // top-2 with lax.top_k tie-breaking (first occurrence wins)
        int i1 = 0;
        for (int e = 1; e < E_NUM; ++e) if (logits[e] > logits[i1]) i1 = e;
        int i2 = -1;
        for (int e = 0; e < E_NUM; ++e) {
            if (e == i1) continue;
            if (i2 < 0 || logits[e] > logits[i2]) i2 = e;
        }
        float m = fmaxf(logits[i1], logits[i2]);
        float ex1 = expf(logits[i1] - m), ex2 = expf(logits[i2] - m);
        float inv = 1.0f / (ex1 + ex2);
        float p1 = ex1 * inv, p2 = ex2 * inv;   // gates for experts i1, i2

        int e_lo = (i1 < i2) ? i1 : i2;
        int e_hi = (i1 < i2) ? i2 : i1;
        float g_lo = (e_lo == i1) ? p1 : p2;
        float g_hi = (e_hi == i1) ? p1 : p2;

        // loss = 0.01 * (cv2(importance) + cv2(load))
        float imp[E_NUM] = {0.f, 0.f, 0.f, 0.f};
        float ld[E_NUM]  = {0.f, 0.f, 0.f, 0.f};
        imp[i1] = p1 * (float)B_ROWS; imp[i2] = p2 * (float)B_ROWS;
        ld[i1] = (float)B_ROWS;       ld[i2] = (float)B_ROWS;
        float loss = 0.01f * (cv_squared4(imp) + cv_squared4(ld));

        wsf[WS_G0] = g_lo;
        wsf[WS_G1] = g_hi;
        ((int*)wsf)[WS_E0] = e_lo;
        ((int*)wsf)[WS_E1] = e_hi;
        loss_out[0] = loss;
        sh_e[0] = e_lo; sh_e[1] = e_hi;
    }
    __syncthreads();

    // W2P[j][k][q] = w2[e_j, k, p(q)] or 0  (KS[e] = 3 + 2e, zero-pad to 9x9)
    for (int idx = tid; idx < 2 * H_DIM * NPAD; idx += blockDim.x) {
        int j = idx / (H_DIM * NPAD);
        int rem = idx % (H_DIM * NPAD);
        int k = rem / NPAD, q = rem % NPAD;
        int e = sh_e[j];
        int ksz = 3 + 2 * e;
        int pad = (9 - ksz) >> 1;
        float val = 0.f;
        if (q < OUT_Q) {
            int R = q / 9, Cc = q % 9;
            if (R >= pad && R < pad + ksz && Cc >= pad && Cc < pad + ksz) {
                int p = (R - pad) * ksz + (Cc - pad);
                val = w2[(e * H_DIM + k) * OUT_Q + p];
            }
        }
        wsf[WS_W2P + idx] = val;
    }
    for (int idx = tid; idx < 2 * NPAD; idx += blockDim.x) {
        int j = idx / NPAD, q = idx % NPAD;
        int e = sh_e[j];
        int ksz = 3 + 2 * e;
        int pad = (9 - ksz) >> 1;
        float val = 0.f;
        if (q < OUT_Q) {
            int R = q / 9, Cc = q % 9;
            if (R >= pad && R < pad + ksz && Cc >= pad && Cc < pad + ksz)
                val = b2[e * OUT_Q + (R - pad) * ksz + (Cc - pad)];
        }
        wsf[WS_B2P + idx] = val;
    }
}

// Main kernel: 4 waves/block, 16 rows per wave, 64 rows per block.
// Per wave: async-stage own 16 A-rows (8 KB) into LDS via
// GLOBAL_LOAD_ASYNC_TO_LDS_B128 (ASYNCcnt), then per expert:
// GEMM1 (16x128 @ 128x32, 64 wmma) -> leaky -> LDS relayout ->
// GEMM2 (16x32 @ 32x96, 48 wmma) -> fused log(g0*exp(z0)+g1*exp(z1)).
__global__ void __launch_bounds__(WAVES_PER_BLOCK * 32)
moe_kernel(const float* __restrict__ Dk,
           const float* __restrict__ w1,
           const float* __restrict__ b1g,
           const float* __restrict__ wsf,
           float* __restrict__ out) {
    __shared__ float Xs[ROWS_PER_BLOCK * C_DIM];            // 32 KB A staging
    __shared__ float Hs[WAVES_PER_BLOCK][16 * H_DIM];       // 8 KB H relayout

    int tid  = threadIdx.x;
    int wave = tid >> 5;
    int lane = tid & 31;
    int half = lane >> 4;    // 0: K pair {0,1}; 1: K pair {2,3}
    int ln   = lane & 15;    // M row (A) / N col (B) within 16
    int rowBase = blockIdx.x * ROWS_PER_BLOCK + wave * 16;

    // ---- stage this wave's 16 A-rows (8 KB) into LDS, 16 B per lane-chunk ----
    float* XsW = &Xs[wave * 16 * C_DIM];
    {
        const float* gsrc = Dk + (size_t)rowBase * C_DIM;   // wave-private rows
#if __has_builtin(__builtin_amdgcn_global_load_async_to_lds_b128)
#pragma unroll
        for (int i = 0; i < 16; ++i) {
            int c = lane + i * 32;                           // 512 chunks / 32 lanes
            __builtin_amdgcn_global_load_async_to_lds_b128(
                (__attribute__((address_space(1))) vsi4*)(const_cast<float*>(gsrc) + c * 4),
                (__attribute__((address_space(3))) vsi4*)(XsW + c * 4),
                0, 0);
        }
        asm volatile("s_wait_asynccnt 0" ::: "memory");
#else
#pragma unroll
        for (int i = 0; i < 16; ++i) {
            int c = lane + i * 32;
            *(float4*)(XsW + c * 4) = *(const float4*)(gsrc + c * 4);
        }
#endif
    }
    // wave-private staging region: no block barrier needed

    const int* wsi = (const int*)wsf;
    float gate0 = wsf[WS_G0];
    float gate1 = wsf[WS_G1];
    int eid[2] = { wsi[WS_E0], wsi[WS_E1] };

    float* hsw = &Hs[wave][0];
    v8f z0[6];  // expert-0 (e_lo) outputs, kept live across the expert loop

    for (int j = 0; j < 2; ++j) {
        int e = eid[j];
        const float* W1 = w1 + (size_t)e * C_DIM * H_DIM;

        // ---------------- GEMM1: H = Dk_tile @ w1[e] ----------------
        v8f accA = {};  // N cols 0..15
        v8f accB = {};  // N cols 16..31
#pragma unroll
        for (int ks = 0; ks < C_DIM / 4; ++ks) {
            int kk = ks * 4 + half * 2;
            v2f a = *(const v2f*)(XsW + ln * C_DIM + kk);   // ds_load_b64
            v2f b0, b1v;
            b0.x  = W1[kk * H_DIM + ln];
            b0.y  = W1[(kk + 1) * H_DIM + ln];
            b1v.x = W1[kk * H_DIM + 16 + ln];
            b1v.y = W1[(kk + 1) * H_DIM + 16 + ln];
            accA = __builtin_amdgcn_wmma_f32_16x16x4_f32(false, a, false, b0,
                                                         (short)0, accA, false, false);
            accB = __builtin_amdgcn_wmma_f32_16x16x4_f32(false, a, false, b1v,
                                                         (short)0, accB, false, false);
        }

        // bias + leaky_relu(0.1), relayout C/D frag -> row-major H in LDS
        float bn0 = b1g[e * H_DIM + ln];
        float bn1 = b1g[e * H_DIM + 16 + ln];
#pragma unroll
        for (int v = 0; v < 8; ++v) {
            int r = v + half * 8;  // C/D layout: VGPR v holds M = v (+8 for hi half)
            float h0 = accA[v] + bn0; h0 = (h0 >= 0.f) ? h0 : 0.1f * h0;
            float h1 = accB[v] + bn1; h1 = (h1 >= 0.f) ? h1 : 0.1f * h1;
            hsw[r * H_DIM + ln]      = h0;
            hsw[r * H_DIM + 16 + ln] = h1;
        }
        // same-wave LDS ops are in-order; no barrier needed (per-wave region)

        // ---------------- GEMM2: Z = H @ W2P[e] (+ b2P) ----------------
        const float* W2P = wsf + WS_W2P + j * (H_DIM * NPAD);
        const float* B2P = wsf + WS_B2P + j * NPAD;
        v8f z[6];
#pragma unroll
        for (int t = 0; t < 6; ++t) z[t] = (v8f){};
#pragma unroll
        for (int ks = 0; ks < H_DIM / 4; ++ks) {
            int kk = ks * 4 + half * 2;
            v2f a2 = *(const v2f*)(hsw + ln * H_DIM + kk);  // ds_load_b64
#pragma unroll
            for (int t = 0; t < 6; ++t) {
                v2f bv;
                bv.x = W2P[kk * NPAD + t * 16 + ln];
                bv.y = W2P[(kk + 1) * NPAD + t * 16 + ln];
                z[t] = __builtin_amdgcn_wmma_f32_16x16x4_f32(false, a2, false, bv,
                                                             (short)0, z[t], false, false);
            }
        }
#pragma unroll
        for (int t = 0; t < 6; ++t) {
            float bq = B2P[t * 16 + ln];
#pragma unroll
            for (int v = 0; v < 8; ++v) z[t][v] += bq;
        }

        if (j == 0) {
#pragma unroll
            for (int t = 0; t < 6; ++t) z0[t] = z[t];
        } else {
            // fused combine: out = log(g_lo*exp(z_lo) + g_hi*exp(z_hi))
#pragma unroll
            for (int t = 0; t < 6; ++t) {
                int q = t * 16 + ln;
                if (q < OUT_Q) {
#pragma unroll
                    for (int v = 0; v < 8; ++v) {
                        int r = v + half * 8;
                        float c = gate0 * expf(z0[t][v]) + gate1 * expf(z[t][v]);
                        out[(size_t)(rowBase + r) * OUT_Q + q] = logf(c);
                    }
                }
            }
        }
    }
}

extern "C" void kernel_launch(void* const* d_in, const int* in_sizes, int n_in,
                              void* d_out, int out_size, void* d_ws, size_t ws_size,
                              hipStream_t stream) {
    const float* x_ds   = (const float*)d_in[0];
    const float* Dk     = (const float*)d_in[1];
    const float* w_gate = (const float*)d_in[2];
    const float* w1     = (const float*)d_in[3];
    const float* b1     = (const float*)d_in[4];
    const float* w2     = (const float*)d_in[5];
    const float* b2     = (const float*)d_in[6];
    float* out = (float*)d_out;
    float* wsf = (float*)d_ws;

    gate_kernel<<<1, 256, 0, stream>>>(x_ds, w_gate, w2, b2, wsf,
                                       out + (size_t)B_ROWS * OUT_Q);
    moe_kernel<<<B_ROWS / ROWS_PER_BLOCK, WAVES_PER_BLOCK * 32, 0, stream>>>(
        Dk, w1, b1, wsf, out);
}